// GraphIsomorphismConv_61160334295755
// MI455X (gfx1250) — compile-verified
//
#include <hip/hip_runtime.h>

#define N_NODE 100000
#define N_EDGE 1600000
#define D_IN   128
#define D_EDGE 16
#define EPS    0.0f
#define TILES  (N_NODE / 16)   /* 6250, exact */

typedef float v2f __attribute__((ext_vector_type(2)));
typedef float v8f __attribute__((ext_vector_type(8)));

// D = A(16x4 f32) * B(4x16 f32) + C(16x16 f32), exact f32 on the WMMA pipe.
static __device__ __forceinline__ v8f wmma4(v2f a, v2f b, v8f c) {
    return __builtin_amdgcn_wmma_f32_16x16x4_f32(
        /*neg_a=*/false, a, /*neg_b=*/false, b,
        /*c_mod=*/(short)0, c, /*reuse_a=*/false, /*reuse_b=*/false);
}

// ---------------------------------------------------------------------------
// Phase 1: acc = (1+eps)*x ; edge_agg = 0
// ---------------------------------------------------------------------------
__global__ __launch_bounds__(256) void gin_init_acc(const float* __restrict__ x,
                                                    float* __restrict__ acc) {
    const int i = blockIdx.x * 256 + threadIdx.x;
    if (i < N_NODE * D_IN) acc[i] = (1.0f + EPS) * x[i];
}

__global__ __launch_bounds__(256) void gin_init_eagg(float* __restrict__ eagg) {
    const int i = blockIdx.x * 256 + threadIdx.x;
    if (i < N_NODE * D_EDGE) eagg[i] = 0.0f;
}

// ---------------------------------------------------------------------------
// Phase 2: edge scatter. One wave32 per edge.
//   acc[v,:]  += x[u,:] * w        (128 f32, float4 per lane, 4 f32 atomics)
//   eagg[v,:] += edge_feature[e,:]*w (16 f32, lanes 0..15)
// acc/eagg (58 MB) are L2-resident (192 MB L2) -> atomics resolve in L2.
// ---------------------------------------------------------------------------
__global__ __launch_bounds__(256) void gin_scatter(
    const long long* __restrict__ el,   // (N_EDGE,2) int64
    const float* __restrict__ ew,       // (N_EDGE,)
    const float* __restrict__ ef,       // (N_EDGE,16)
    const float* __restrict__ x,        // (N_NODE,128)
    float* __restrict__ acc,
    float* __restrict__ eagg) {
    const int e = (blockIdx.x * 256 + threadIdx.x) >> 5;
    if (e >= N_EDGE) return;
    const int lane = threadIdx.x & 31;

    const int u = (int)el[2 * (long long)e];
    const int v = (int)el[2 * (long long)e + 1];
    const float w = ew[e];

    const float4 xv = *(const float4*)(x + (size_t)u * D_IN + lane * 4);
    float* dst = acc + (size_t)v * D_IN + lane * 4;
    unsafeAtomicAdd(dst + 0, xv.x * w);   // global_atomic_add_f32 (no return)
    unsafeAtomicAdd(dst + 1, xv.y * w);
    unsafeAtomicAdd(dst + 2, xv.z * w);
    unsafeAtomicAdd(dst + 3, xv.w * w);

    if (lane < D_EDGE) {
        const float f = ef[(size_t)e * D_EDGE + lane];
        unsafeAtomicAdd(eagg + (size_t)v * D_EDGE + lane, f * w);
    }
}

// ---------------------------------------------------------------------------
// Phase 3: fused per-node network, one wave per 16-node tile.
//   H  = acc + b_edge + eagg @ W_edge        (WMMA f32 16x16x4, K=16)
//   h1 = relu(H @ W1 + b1)                   (K=128 -> 32 K-steps x 8 N-tiles)
//   out= relu(h1 @ W2 + b2)
// Fragment layouts (ISA 7.12.2):
//   A (16x4):  lane l -> M=l%16 ; vgpr j -> K = j + 2*(l/16)
//   B (4x16):  lane l -> N=l%16 ; vgpr j -> K = j + 2*(l/16)   (column-major dual)
//   C (16x16): vgpr r, lane l -> M = r + 8*(l/16), N = l%16
// LDS tile stride 132 -> bank-conflict-free C-layout stores and A-layout loads.
// ---------------------------------------------------------------------------
#define LSTR 132

__global__ __launch_bounds__(128) void gin_mlp(
    const float* __restrict__ acc,
    const float* __restrict__ eagg,
    const float* __restrict__ W_edge, const float* __restrict__ b_edge,
    const float* __restrict__ W1, const float* __restrict__ b1,
    const float* __restrict__ W2, const float* __restrict__ b2,
    float* __restrict__ out) {
    __shared__ float lds[4][16 * LSTR];   // 33,792 B: per-wave 16x128 tile

    const int wave = threadIdx.x >> 5;
    const int lane = threadIdx.x & 31;
    const int tile = blockIdx.x * 4 + wave;
    if (tile >= TILES) return;            // uniform per wave

    const int m16 = lane & 15;
    const int kh  = lane >> 4;            // K half (0/1)
    float* hl = lds[wave];

    // ---- Stage 1: H = acc + b_edge + eagg @ W_edge  -> LDS ----
    const int node_m = tile * 16 + m16;
    v2f ae[4];
#pragma unroll
    for (int kk = 0; kk < 4; ++kk) {
        const float* p = eagg + (size_t)node_m * D_EDGE + kk * 4 + 2 * kh;
        ae[kk].x = p[0];
        ae[kk].y = p[1];
    }
#pragma unroll
    for (int nt = 0; nt < 8; ++nt) {
        const int n = nt * 16 + m16;
        const float be = b_edge[n];
        v8f c;
#pragma unroll
        for (int r = 0; r < 8; ++r)
            c[r] = acc[(size_t)(tile * 16 + r + 8 * kh) * D_IN + n] + be;
#pragma unroll
        for (int kk = 0; kk < 4; ++kk) {
            const int k0 = kk * 4 + 2 * kh;
            v2f b;
            b.x = W_edge[k0 * D_IN + n];
            b.y = W_edge[(k0 + 1) * D_IN + n];
            c = wmma4(ae[kk], b, c);
        }
#pragma unroll
        for (int r = 0; r < 8; ++r)
            hl[(r + 8 * kh) * LSTR + n] = c[r];
    }
    asm volatile("s_wait_dscnt 0" ::: "memory");

    // ---- Stage 2: h1 = relu(H @ W1 + b1) -> LDS ----
    v2f a[32];
#pragma unroll
    for (int kk = 0; kk < 32; ++kk) {
        const int k0 = kk * 4 + 2 * kh;
        a[kk].x = hl[m16 * LSTR + k0];
        a[kk].y = hl[m16 * LSTR + k0 + 1];
    }
    asm volatile("s_wait_dscnt 0" ::: "memory");
    for (int nt = 0; nt < 8; ++nt) {
        const int n = nt * 16 + m16;
        const float bb = b1[n];
        v8f c;
#pragma unroll
        for (int r = 0; r < 8; ++r) c[r] = bb;
#pragma unroll
        for (int kk = 0; kk < 32; ++kk) {
            const int k0 = kk * 4 + 2 * kh;
            v2f b;
            b.x = W1[k0 * D_IN + n];
            b.y = W1[(k0 + 1) * D_IN + n];
            c = wmma4(a[kk], b, c);
        }
#pragma unroll
        for (int r = 0; r < 8; ++r)
            hl[(r + 8 * kh) * LSTR + n] = fmaxf(c[r], 0.0f);
    }
    asm volatile("s_wait_dscnt 0" ::: "memory");

    // ---- Stage 3: out = relu(h1 @ W2 + b2) -> global ----
#pragma unroll
    for (int kk = 0; kk < 32; ++kk) {
        const int k0 = kk * 4 + 2 * kh;
        a[kk].x = hl[m16 * LSTR + k0];
        a[kk].y = hl[m16 * LSTR + k0 + 1];
    }
    for (int nt = 0; nt < 8; ++nt) {
        const int n = nt * 16 + m16;
        const float bb = b2[n];
        v8f c;
#pragma unroll
        for (int r = 0; r < 8; ++r) c[r] = bb;
#pragma unroll
        for (int kk = 0; kk < 32; ++kk) {
            const int k0 = kk * 4 + 2 * kh;
            v2f b;
            b.x = W2[k0 * D_IN + n];
            b.y = W2[(k0 + 1) * D_IN + n];
            c = wmma4(a[kk], b, c);
        }
#pragma unroll
        for (int r = 0; r < 8; ++r)
            out[(size_t)(tile * 16 + r + 8 * kh) * D_IN + n] = fmaxf(c[r], 0.0f);
    }
}

// ---------------------------------------------------------------------------
extern "C" void kernel_launch(void* const* d_in, const int* in_sizes, int n_in,
                              void* d_out, int out_size, void* d_ws, size_t ws_size,
                              hipStream_t stream) {
    const long long* el     = (const long long*)d_in[0];
    const float*     ew     = (const float*)d_in[1];
    const float*     ef     = (const float*)d_in[2];
    /* d_in[3] = num_node (compile-time constant here) */
    const float*     x      = (const float*)d_in[4];
    const float*     W_edge = (const float*)d_in[5];
    const float*     b_edge = (const float*)d_in[6];
    const float*     W1     = (const float*)d_in[7];
    const float*     b1     = (const float*)d_in[8];
    const float*     W2     = (const float*)d_in[9];
    const float*     b2     = (const float*)d_in[10];
    float*           outp   = (float*)d_out;

    float* acc  = (float*)d_ws;                       // 100000*128 f32 = 51.2 MB
    float* eagg = acc + (size_t)N_NODE * D_IN;        // 100000*16  f32 =  6.4 MB

    gin_init_acc<<<(N_NODE * D_IN + 255) / 256, 256, 0, stream>>>(x, acc);
    gin_init_eagg<<<(N_NODE * D_EDGE + 255) / 256, 256, 0, stream>>>(eagg);

    // one wave32 per edge: N_EDGE waves / 8 waves per block
    gin_scatter<<<(N_EDGE * 32 + 255) / 256, 256, 0, stream>>>(el, ew, ef, x, acc, eagg);

    gin_mlp<<<(TILES + 3) / 4, 128, 0, stream>>>(acc, eagg, W_edge, b_edge,
                                                 W1, b1, W2, b2, outp);
}